// QPSolver_82695300317767
// MI455X (gfx1250) — compile-verified
//
#include <hip/hip_runtime.h>

// ---------------------------------------------------------------------------
// QP fixed-point solver for MI455X (gfx1250, wave32, WMMA).
//   N=32, M=64, BS=512, iters=1000, ALPHA=BETA=1
// Hot loop uses V_WMMA_F32_16X16X4_F32 (fp32 in/out) with all constant
// operands pinned in VGPRs; X exchanged via LDS with 2 barriers/iter.
// ---------------------------------------------------------------------------

typedef __attribute__((ext_vector_type(2))) float v2f;
typedef __attribute__((ext_vector_type(8))) float v8f;

// Workspace layout (float offsets)
constexpr int WS_PINV  = 0;                    // 32x32
constexpr int WS_HINV  = 1024;                 // 32x64  (pinv(H), row-major n,k)
constexpr int WS_A     = 3072;                 // 128x128 iteration matrix A (row-major)
constexpr int WS_S     = 19456;                // scalar s = BETA / lam_max
constexpr int WS_PINVQ = 19472;                // 512x32
constexpr int WS_B     = WS_PINVQ + 512 * 32;  // 512x128 bias [mu, -2mu]

constexpr int BS = 512, NN = 32, MM = 64, TWO_M = 128, KDIM = 1001;
constexpr int LSTR = 132;  // LDS row stride (floats), padded

// ---------------------------------------------------------------------------
// Setup: Pinv, Ginv -> Hinv = pinv(H), D = H Pinv H^T, power iteration for
// lam_max, assemble A.  Single workgroup; tiny cost.
// ---------------------------------------------------------------------------
__global__ __launch_bounds__(256) void qp_setup(const float* __restrict__ P,
                                                const float* __restrict__ H,
                                                float* __restrict__ ws) {
  __shared__ float Ap[32][66];  // [P | I]
  __shared__ float Ag[32][66];  // [H^T H | I]
  __shared__ float Tm[32][66];  // Pinv @ H^T
  __shared__ float Dm[64][66];  // D = H Pinv H^T
  __shared__ float colP[32], colG[32];
  __shared__ float vvec[64], wvec[64];
  __shared__ float lam;
  const int tid = threadIdx.x;

  for (int idx = tid; idx < 32 * 64; idx += 256) {
    int r = idx >> 6, c = idx & 63;
    Ap[r][c] = (c < 32) ? P[r * 32 + c] : ((c - 32 == r) ? 1.f : 0.f);
    float g;
    if (c < 32) {
      float acc = 0.f;
      for (int k = 0; k < 64; ++k) acc += H[k * 32 + r] * H[k * 32 + c];
      g = acc;
    } else {
      g = (c - 32 == r) ? 1.f : 0.f;
    }
    Ag[r][c] = g;
  }
  __syncthreads();

  // Gauss-Jordan on both SPD matrices simultaneously (no pivoting needed)
  for (int piv = 0; piv < 32; ++piv) {
    if (tid < 32) { colP[tid] = Ap[tid][piv]; colG[tid] = Ag[tid][piv]; }
    __syncthreads();
    float ipv = 1.f / colP[piv];
    float igv = 1.f / colG[piv];
    if (tid < 64) { Ap[piv][tid] *= ipv; Ag[piv][tid] *= igv; }
    __syncthreads();
    for (int idx = tid; idx < 32 * 64; idx += 256) {
      int r = idx >> 6, c = idx & 63;
      if (r != piv) {
        Ap[r][c] -= colP[r] * Ap[piv][c];
        Ag[r][c] -= colG[r] * Ag[piv][c];
      }
    }
    __syncthreads();
  }

  for (int idx = tid; idx < 32 * 32; idx += 256)
    ws[WS_PINV + idx] = Ap[idx >> 5][32 + (idx & 31)];
  // Hinv = Ginv @ H^T ; Tm = Pinv @ H^T
  for (int idx = tid; idx < 32 * 64; idx += 256) {
    int n = idx >> 6, j = idx & 63;
    float a1 = 0.f, a2 = 0.f;
    for (int c = 0; c < 32; ++c) {
      float h = H[j * 32 + c];
      a1 += Ag[n][32 + c] * h;
      a2 += Ap[n][32 + c] * h;
    }
    ws[WS_HINV + idx] = a1;
    Tm[n][j] = a2;
  }
  __syncthreads();
  // D = H @ Tm
  for (int idx = tid; idx < 64 * 64; idx += 256) {
    int i = idx >> 6, j = idx & 63;
    float acc = 0.f;
    for (int n = 0; n < 32; ++n) acc += H[i * 32 + n] * Tm[n][j];
    Dm[i][j] = acc;
  }
  if (tid < 64) vvec[tid] = 1.f;
  __syncthreads();
  // Power iteration for lam_max(D)
  for (int it = 0; it < 96; ++it) {
    if (tid < 64) {
      float acc = 0.f;
      for (int j = 0; j < 64; ++j) acc += Dm[tid][j] * vvec[j];
      wvec[tid] = acc;
    }
    __syncthreads();
    if (tid == 0) {
      float nrm = 0.f;
      for (int j = 0; j < 64; ++j) nrm += wvec[j] * wvec[j];
      lam = sqrtf(nrm);
    }
    __syncthreads();
    if (tid < 64) vvec[tid] = wvec[tid] / lam;
    __syncthreads();
  }
  float s = 1.0f / lam;  // BETA / lam_max
  if (tid == 0) ws[WS_S] = s;
  // A = [[ sD, sI ], [ -2sD + I, (1-2s)I ]]
  for (int idx = tid; idx < 128 * 128; idx += 256) {
    int r = idx >> 7, c = idx & 127;
    float v;
    if (r < 64) {
      v = (c < 64) ? s * Dm[r][c] : ((c - 64 == r) ? s : 0.f);
    } else {
      int rr = r - 64;
      if (c < 64) v = -2.f * s * Dm[rr][c] + ((rr == c) ? 1.f : 0.f);
      else        v = (r == c) ? (1.f - 2.f * s) : 0.f;
    }
    ws[WS_A + idx] = v;
  }
}

// Pinv_q = q @ Pinv  (Pinv symmetric)
__global__ __launch_bounds__(256) void qp_pinvq(const float* __restrict__ q,
                                                float* __restrict__ ws) {
  int idx = blockIdx.x * 256 + threadIdx.x;
  if (idx >= BS * NN) return;
  int r = idx >> 5, n = idx & 31;
  float acc = 0.f;
  for (int c = 0; c < 32; ++c) acc += q[r * 32 + c] * ws[WS_PINV + c * 32 + n];
  ws[WS_PINVQ + idx] = acc;
}

// Bbias = [mu, -2mu],  mu = s * (H @ Pinv_q^T)^T - s*b
__global__ __launch_bounds__(256) void qp_bbias(const float* __restrict__ b,
                                                const float* __restrict__ H,
                                                float* __restrict__ ws) {
  int idx = blockIdx.x * 256 + threadIdx.x;
  if (idx >= BS * MM) return;
  int r = idx >> 6, j = idx & 63;
  float s = ws[WS_S];
  float acc = 0.f;
  for (int n = 0; n < 32; ++n) acc += H[j * 32 + n] * ws[WS_PINVQ + r * 32 + n];
  float mu = s * (acc - b[idx]);
  ws[WS_B + r * 128 + j] = mu;
  ws[WS_B + r * 128 + 64 + j] = -2.f * mu;
}

// ---------------------------------------------------------------------------
// Main loop: 32 blocks x 256 threads (8 waves).  Block owns 16 batch rows;
// wave w owns X columns [16w,16w+16).  Per iteration: 32 chained
// V_WMMA_F32_16X16X4_F32 (K=128) + 16 for the primal projection.
// ---------------------------------------------------------------------------
__global__ __launch_bounds__(256) void qp_iterate(const float* __restrict__ b,
                                                  const float* __restrict__ ws,
                                                  const int* __restrict__ iters_p,
                                                  float* __restrict__ out) {
  __shared__ float X[16 * LSTR];
  const int tid  = threadIdx.x;
  const int wave = tid >> 5;
  const int lane = tid & 31;
  const int half = lane >> 4;   // selects K-pair / M-half per WMMA layout
  const int nl   = lane & 15;
  const int r0   = blockIdx.x * 16;
  const int iters = *iters_p;

  const float* Am = ws + WS_A;
  const float* Hi = ws + WS_HINV;
  const float* Bb = ws + WS_B;
  float* Xs = out;
  float* Pr = out + (size_t)BS * KDIM * TWO_M;

  const int ncol = wave * 16 + nl;        // global X column of this wave's tile
  const int pcol = (wave & 1) * 16 + nl;  // primal column (tile mirrored on all waves)

  // Preload constant B-operands into VGPRs for the whole run.
  // B-frag layout assumed: v[0] -> K = 4*kt + 2*half, v[1] -> K+1, N = nl.
  v2f afr[32];
#pragma unroll
  for (int kt = 0; kt < 32; ++kt) {
    const float* p = Am + (size_t)ncol * 128 + kt * 4 + half * 2;  // A^T[k][n] = A[n][k]
    afr[kt][0] = p[0]; afr[kt][1] = p[1];
  }
  v2f hfr[16];
#pragma unroll
  for (int kt = 0; kt < 16; ++kt) {
    const float* p = Hi + (size_t)pcol * 64 + kt * 4 + half * 2;   // Hinv^T[k][n] = Hinv[n][k]
    hfr[kt][0] = p[0]; hfr[kt][1] = p[1];
  }
  // C/D layout: VGPR v -> (M = v + 8*half, N = nl)
  v8f bias;
#pragma unroll
  for (int v = 0; v < 8; ++v)
    bias[v] = Bb[(size_t)(r0 + v + half * 8) * 128 + ncol];

  // Stage b into the dual region of LDS to build pbias = -(b @ Hinv^T).
  for (int idx = tid; idx < 16 * 64; idx += 256) {
    int rr = idx >> 6, cc = idx & 63;
    X[rr * LSTR + 64 + cc] = b[(size_t)(r0 + rr) * 64 + cc];
  }
  __syncthreads();

  v8f pb;
#pragma unroll
  for (int v = 0; v < 8; ++v) pb[v] = 0.f;
#pragma unroll
  for (int kt = 0; kt < 16; ++kt) {
    const float* xp = &X[nl * LSTR + 64 + kt * 4 + half * 2];
    v2f xf; xf[0] = xp[0]; xf[1] = xp[1];
    pb = __builtin_amdgcn_wmma_f32_16x16x4_f32(false, xf, false, hfr[kt],
                                               (short)0, pb, false, false);
  }
#pragma unroll
  for (int v = 0; v < 8; ++v) pb[v] = -pb[v];

  // k = 0: Xs row is all zeros, primal row is pb.
#pragma unroll
  for (int v = 0; v < 8; ++v) {
    size_t row = (size_t)(r0 + v + half * 8);
    Xs[row * KDIM * TWO_M + ncol] = 0.f;
    if (wave < 2) Pr[row * KDIM * NN + pcol] = pb[v];
  }
  __syncthreads();
  for (int idx = tid; idx < 16 * LSTR; idx += 256) X[idx] = 0.f;
  __syncthreads();

  for (int k = 1; k <= iters; ++k) {
    // newX tile = X(16x128) @ A^T(128x16) + bias : 32 chained K=4 WMMAs
    v8f acc = bias;
#pragma unroll
    for (int kt = 0; kt < 32; ++kt) {
      const float* xp = &X[nl * LSTR + kt * 4 + half * 2];  // ds_load_b64
      v2f xf; xf[0] = xp[0]; xf[1] = xp[1];
      acc = __builtin_amdgcn_wmma_f32_16x16x4_f32(false, xf, false, afr[kt],
                                                  (short)0, acc, false, false);
    }
    if (wave >= 4) {  // ReLU on dual half (columns 64..127)
#pragma unroll
      for (int v = 0; v < 8; ++v) acc[v] = fmaxf(acc[v], 0.f);
    }
    __syncthreads();  // everyone done reading old X
#pragma unroll
    for (int v = 0; v < 8; ++v) {
      X[(v + half * 8) * LSTR + ncol] = acc[v];
      Xs[((size_t)(r0 + v + half * 8) * KDIM + k) * TWO_M + ncol] = acc[v];
    }
    __syncthreads();  // new X visible

    // primal = X_dual @ Hinv^T + pbias.  All waves compute (EXEC stays all-1s
    // for WMMA); waves 0-1 hold the two distinct tiles and store them.
    v8f p = pb;
#pragma unroll
    for (int kt = 0; kt < 16; ++kt) {
      const float* xp = &X[nl * LSTR + 64 + kt * 4 + half * 2];
      v2f xf; xf[0] = xp[0]; xf[1] = xp[1];
      p = __builtin_amdgcn_wmma_f32_16x16x4_f32(false, xf, false, hfr[kt],
                                                (short)0, p, false, false);
    }
    if (wave < 2) {
#pragma unroll
      for (int v = 0; v < 8; ++v)
        Pr[((size_t)(r0 + v + half * 8) * KDIM + k) * NN + pcol] = p[v];
    }
  }
}

extern "C" void kernel_launch(void* const* d_in, const int* in_sizes, int n_in,
                              void* d_out, int out_size, void* d_ws, size_t ws_size,
                              hipStream_t stream) {
  (void)in_sizes; (void)n_in; (void)out_size; (void)ws_size;
  const float* q = (const float*)d_in[0];
  const float* b = (const float*)d_in[1];
  const float* P = (const float*)d_in[2];
  const float* H = (const float*)d_in[3];
  const int* iters = (const int*)d_in[4];
  float* ws = (float*)d_ws;
  float* out = (float*)d_out;

  qp_setup<<<1, 256, 0, stream>>>(P, H, ws);
  qp_pinvq<<<(BS * NN + 255) / 256, 256, 0, stream>>>(q, ws);
  qp_bbias<<<(BS * MM + 255) / 256, 256, 0, stream>>>(b, H, ws);
  qp_iterate<<<BS / 16, 256, 0, stream>>>(b, ws, iters, out);
}